// RPEAttention_81879256531613
// MI455X (gfx1250) — compile-verified
//
#include <hip/hip_runtime.h>
#include <hip/hip_bf16.h>
#include <math.h>

// ---------------------------------------------------------------------------
// RPE attention for MI455X (gfx1250, wave32).
// fp32 end-to-end (matches reference precision); all GEMM-shaped work runs on
// V_WMMA_F32_16X16X4_F32. Op is HBM-bound (~0.6 GB traffic -> ~26us floor at
// 23.3 TB/s). GEMM K-tiles are streamed memory->LDS with
// GLOBAL_LOAD_ASYNC_TO_LDS_B128 (ASYNCcnt) and double-buffered so global
// fetch overlaps WMMA; falls back to the sync VGPR-relay path if the async
// builtins are unavailable in this toolchain.
// ---------------------------------------------------------------------------

typedef float v2f __attribute__((ext_vector_type(2)));
typedef float v8f __attribute__((ext_vector_type(8)));
typedef int b128_t __attribute__((vector_size(16)));            // B128 payload
typedef __attribute__((address_space(3))) b128_t lds_b128_t;    // LDS-side

__device__ __forceinline__ v8f wmma_f32_16x16x4(v2f a, v2f b, v8f c) {
  // 8 args: (neg_a, A, neg_b, B, c_mod, C, reuse_a, reuse_b)
  return __builtin_amdgcn_wmma_f32_16x16x4_f32(false, a, false, b, (short)0, c,
                                               false, false);
}

#if defined(__has_builtin)
#if __has_builtin(__builtin_amdgcn_global_load_async_to_lds_b128) && \
    __has_builtin(__builtin_amdgcn_s_wait_asynccnt)
#define HAVE_ASYNC_LDS 1
#endif
#endif
#ifndef HAVE_ASYNC_LDS
#define HAVE_ASYNC_LDS 0
#endif

#define NEG_INF (-__builtin_inff())

// Problem constants
#define BB   2
#define DD   512
#define CC   512
#define TT   32
#define TDIM 1024
#define HH   8
#define FF   64
#define SCALE 0.125f   // F^-0.5

// LDS tile row stride: 36 floats = 144B -> 16B aligned rows (required for
// async B128 LDS writes) and bank-conflict-free fragment gathers
// (gcd(36,64)=4 => 16 distinct banks over a 16-lane half-wave).
#define LSTR 36

// ---------------------------------------------------------------------------
// GroupNorm + transpose: x[bd, C, T] -> xt[bd, T, C]
// Group g occupies contiguous floats [g*512, (g+1)*512) within a bd slice.
// ---------------------------------------------------------------------------
__global__ __launch_bounds__(256) void groupnorm_kernel(
    const float* __restrict__ x, const float* __restrict__ w,
    const float* __restrict__ bvec, float* __restrict__ xt) {
  __shared__ float s1[256], s2[256];
  __shared__ float mu[32], rs[32];
  const int bd = blockIdx.x;
  const int tid = threadIdx.x;
  const float* xb = x + (size_t)bd * (CC * TT);

  const int g = tid >> 3, sub = tid & 7;
  float sum = 0.f, sq = 0.f;
  for (int j = 0; j < 64; ++j) {
    float v = xb[g * 512 + sub + j * 8];
    sum += v; sq += v * v;
  }
  s1[tid] = sum; s2[tid] = sq;
  __syncthreads();
  if (tid < 32) {
    float S = 0.f, Q = 0.f;
    for (int j = 0; j < 8; ++j) { S += s1[tid * 8 + j]; Q += s2[tid * 8 + j]; }
    float m = S * (1.f / 512.f);
    float var = Q * (1.f / 512.f) - m * m;
    mu[tid] = m;
    rs[tid] = rsqrtf(var + 1e-5f);
  }
  __syncthreads();
  for (int i = tid; i < CC * TT; i += 256) {
    int c = i >> 5, t = i & 31, gg = c >> 4;
    float v = (xb[i] - mu[gg]) * rs[gg] * w[c] + bvec[c];
    xt[((size_t)bd * TT + t) * CC + c] = v;
  }
}

// ---------------------------------------------------------------------------
// Generic fp32 WMMA GEMM: C[M,N] = A[M,K] @ W[N,K]^T + bias[N]
// mode 0: plain row-major store
// mode 1: val += resid[m*N+n]; store transposed: out[(m/T)*N*T + n*T + m%T]
// Block tile 64x64, 8 waves, 2 subtiles/wave, K staged 32 wide via LDS with
// async double buffering. Requires M%64==0, N%64==0, K%32==0 (true here).
// ---------------------------------------------------------------------------
#if HAVE_ASYNC_LDS
// One wave issues its share of a 64x32 tile: rows [wave*8, wave*8+8), two
// B128 instructions (each: 32 lanes x 16B = 4 rows of 128B).
__device__ __forceinline__ void async_tile_issue(
    const float* __restrict__ gbase,  // tile origin in global (row stride K)
    float (*lds)[LSTR], int K, int wave, int lane) {
  const int seg = lane & 7;                    // 16B segment within a row
  const int r0 = (wave << 3) + ((lane >> 3) & 3);
  const float* g0 = gbase + (size_t)r0 * K + seg * 4;
  const float* g1 = gbase + (size_t)(r0 + 4) * K + seg * 4;
  __builtin_amdgcn_global_load_async_to_lds_b128(
      (b128_t*)g0,
      (lds_b128_t*)(__attribute__((address_space(3))) void*)&lds[r0][seg * 4],
      0, 0);
  __builtin_amdgcn_global_load_async_to_lds_b128(
      (b128_t*)g1,
      (lds_b128_t*)(__attribute__((address_space(3))) void*)&lds[r0 + 4][seg * 4],
      0, 0);
}
#define GEMM_NBUF 2
#else
#define GEMM_NBUF 1
#endif

__global__ __launch_bounds__(256) void gemm_wmma_kernel(
    const float* __restrict__ A, const float* __restrict__ W,
    const float* __restrict__ bias, float* __restrict__ Cout,
    int M, int N, int K,
    const float* __restrict__ resid, int mode, int Tdim) {
  __shared__ float As[GEMM_NBUF][64][LSTR];
  __shared__ float Ws[GEMM_NBUF][64][LSTR];
  const int m0 = blockIdx.y * 64;
  const int n0 = blockIdx.x * 64;
  const int tid = threadIdx.x;
  const int wave = tid >> 5;
  const int lane = tid & 31;

  const int st0 = wave * 2, st1 = wave * 2 + 1;
  const int sm0 = (st0 >> 2) * 16, sn0 = (st0 & 3) * 16;
  const int sm1 = (st1 >> 2) * 16, sn1 = (st1 & 3) * 16;

  v8f acc0 = {0.f, 0.f, 0.f, 0.f, 0.f, 0.f, 0.f, 0.f};
  v8f acc1 = acc0;

  const int kl = (lane < 16) ? 0 : 2;
  const int ml = lane & 15;

#if HAVE_ASYNC_LDS
  const float* Abase = A + (size_t)m0 * K;
  const float* Wbase = W + (size_t)n0 * K;
  const int ntiles = K >> 5;
  // prologue: fetch tile 0 into buffer 0
  async_tile_issue(Abase, As[0], K, wave, lane);
  async_tile_issue(Wbase, Ws[0], K, wave, lane);
  for (int it = 0; it < ntiles; ++it) {
    const int buf = it & 1;
    if (it + 1 < ntiles) {
      // fetch next tile into the other buffer, then wait for tile `it`
      // (async loads complete in order; <=4 outstanding => previous 4 done)
      async_tile_issue(Abase + (it + 1) * 32, As[buf ^ 1], K, wave, lane);
      async_tile_issue(Wbase + (it + 1) * 32, Ws[buf ^ 1], K, wave, lane);
      __builtin_amdgcn_s_wait_asynccnt(4);
    } else {
      __builtin_amdgcn_s_wait_asynccnt(0);
    }
    __syncthreads();  // tile `it` visible to all waves
    for (int kk = 0; kk < 32; kk += 4) {
      v2f a0, b0, a1, b1;
      a0.x = As[buf][sm0 + ml][kk + kl]; a0.y = As[buf][sm0 + ml][kk + kl + 1];
      b0.x = Ws[buf][sn0 + ml][kk + kl]; b0.y = Ws[buf][sn0 + ml][kk + kl + 1];
      acc0 = wmma_f32_16x16x4(a0, b0, acc0);
      a1.x = As[buf][sm1 + ml][kk + kl]; a1.y = As[buf][sm1 + ml][kk + kl + 1];
      b1.x = Ws[buf][sn1 + ml][kk + kl]; b1.y = Ws[buf][sn1 + ml][kk + kl + 1];
      acc1 = wmma_f32_16x16x4(a1, b1, acc1);
    }
    __syncthreads();  // all waves done reading `buf` before it is refilled
  }
#else
  const int c = tid & 31;   // k within tile
  const int rb = tid >> 5;  // row base (0..7)
  for (int k0 = 0; k0 < K; k0 += 32) {
    for (int i = 0; i < 8; ++i) {
      int r2 = rb + i * 8;
      As[0][r2][c] = A[(size_t)(m0 + r2) * K + k0 + c];
      Ws[0][r2][c] = W[(size_t)(n0 + r2) * K + k0 + c];
      if (k0 + 32 < K) {  // prefetch next K tile -> global_prefetch_b8
        __builtin_prefetch(&A[(size_t)(m0 + r2) * K + k0 + 32 + c], 0, 1);
        __builtin_prefetch(&W[(size_t)(n0 + r2) * K + k0 + 32 + c], 0, 1);
      }
    }
    __syncthreads();
    for (int kk = 0; kk < 32; kk += 4) {
      v2f a0, b0, a1, b1;
      a0.x = As[0][sm0 + ml][kk + kl]; a0.y = As[0][sm0 + ml][kk + kl + 1];
      b0.x = Ws[0][sn0 + ml][kk + kl]; b0.y = Ws[0][sn0 + ml][kk + kl + 1];
      acc0 = wmma_f32_16x16x4(a0, b0, acc0);
      a1.x = As[0][sm1 + ml][kk + kl]; a1.y = As[0][sm1 + ml][kk + kl + 1];
      b1.x = Ws[0][sn1 + ml][kk + kl]; b1.y = Ws[0][sn1 + ml][kk + kl + 1];
      acc1 = wmma_f32_16x16x4(a1, b1, acc1);
    }
    __syncthreads();
  }
#endif

  const int mrow = (lane >= 16) ? 8 : 0;
  for (int pass = 0; pass < 2; ++pass) {
    v8f acc = pass ? acc1 : acc0;
    int sm = pass ? sm1 : sm0;
    int sn = pass ? sn1 : sn0;
    for (int r = 0; r < 8; ++r) {
      int m = m0 + sm + r + mrow;
      int n = n0 + sn + ml;
      float val = acc[r] + bias[n];
      if (mode == 0) {
        Cout[(size_t)m * N + n] = val;
      } else {
        val += resid[(size_t)m * N + n];
        int bd = m / Tdim;
        int t = m - bd * Tdim;
        Cout[((size_t)bd * N + n) * Tdim + t] = val;
      }
    }
  }
}

// ---------------------------------------------------------------------------
// RPE features + silu: silu_emb[b,t,s,c] = silu(tproj[b,t,c] + feats@dw^T + db)
// ---------------------------------------------------------------------------
__global__ __launch_bounds__(256) void rpe_emb_kernel(
    const float* __restrict__ tproj, const float* __restrict__ dw,
    const float* __restrict__ db, const int* __restrict__ fi,
    float* __restrict__ silu_emb) {
  int idx = blockIdx.x * 256 + threadIdx.x;
  if (idx >= BB * TT * TT * CC) return;
  int c = idx & 511;
  int s = (idx >> 9) & 31;
  int t = (idx >> 14) & 31;
  int b = idx >> 19;
  int d = fi[b * TT + t] - fi[b * TT + s];
  float df = (float)d;
  float f0 = log1pf(fmaxf(df, 0.f));
  float f1 = log1pf(fmaxf(-df, 0.f));
  float f2 = (d == 0) ? 1.f : 0.f;
  float e = tproj[((size_t)b * TT + t) * CC + c] + f0 * dw[c * 3 + 0] +
            f1 * dw[c * 3 + 1] + f2 * dw[c * 3 + 2] + db[c];
  silu_emb[idx] = e / (1.f + __expf(-e));
}

// ---------------------------------------------------------------------------
// Attention core, one block (4 waves) per (b,d,h).
// WMMA for q@k^T (K=64) and attn@v (K=32); VALU float4 dots for RPE terms
// (R tensors total 12MB -> L2-resident on the 192MB L2).
// ---------------------------------------------------------------------------
__global__ __launch_bounds__(128) void attn_kernel(
    const float* __restrict__ qkv,  // [B*D, T, 3C]
    const float* __restrict__ Rk, const float* __restrict__ Rq,
    const float* __restrict__ Rv, const int* __restrict__ amask,
    float* __restrict__ out) {      // [B*D, T, C]
  const int h = blockIdx.x & 7;
  const int d = (blockIdx.x >> 3) & 511;
  const int b = blockIdx.x >> 12;
  const size_t bd = (size_t)b * DD + d;

  __shared__ float qs[32][68];
  __shared__ float ks[32][68];
  __shared__ float vs[32][68];
  __shared__ float sc[32][33];
  __shared__ float os[32][68];

  const int tid = threadIdx.x;
  const int wave = tid >> 5;
  const int lane = tid & 31;
  const int kl = (lane < 16) ? 0 : 2;
  const int ml = lane & 15;
  const int mrow = (lane >= 16) ? 8 : 0;

  // Load q (pre-scaled), k, v tiles: 16 float4 per row.
  for (int i = tid; i < TT * 16; i += 128) {
    int t = i >> 4;
    int f4 = (i & 15) * 4;
    const float* base = qkv + (bd * TT + t) * (3 * CC) + h * FF + f4;
    float4 q4 = *(const float4*)(base);
    float4 k4 = *(const float4*)(base + CC);
    float4 v4 = *(const float4*)(base + 2 * CC);
    qs[t][f4 + 0] = q4.x * SCALE; qs[t][f4 + 1] = q4.y * SCALE;
    qs[t][f4 + 2] = q4.z * SCALE; qs[t][f4 + 3] = q4.w * SCALE;
    ks[t][f4 + 0] = k4.x; ks[t][f4 + 1] = k4.y;
    ks[t][f4 + 2] = k4.z; ks[t][f4 + 3] = k4.w;
    vs[t][f4 + 0] = v4.x; vs[t][f4 + 1] = v4.y;
    vs[t][f4 + 2] = v4.z; vs[t][f4 + 3] = v4.w;
  }
  __syncthreads();

  // scores = q @ k^T : four 16x16 tiles, one per wave
  {
    int sm = (wave >> 1) * 16, sn = (wave & 1) * 16;
    v8f acc = {0.f, 0.f, 0.f, 0.f, 0.f, 0.f, 0.f, 0.f};
    for (int kk = 0; kk < FF; kk += 4) {
      v2f a, bf;
      a.x = qs[sm + ml][kk + kl]; a.y = qs[sm + ml][kk + kl + 1];
      bf.x = ks[sn + ml][kk + kl]; bf.y = ks[sn + ml][kk + kl + 1];
      acc = wmma_f32_16x16x4(a, bf, acc);
    }
    for (int r = 0; r < 8; ++r) sc[sm + r + mrow][sn + ml] = acc[r];
  }
  __syncthreads();

  // RPE biases (q.Rk + scale*k.Rq) and mask (allowed iff m_t == m_s)
  for (int i = tid; i < TT * TT; i += 128) {
    int t = i >> 5, s = i & 31;
    const float4* rk4 =
        (const float4*)(Rk + (((size_t)b * TT + t) * TT + s) * CC + h * FF);
    const float4* rq4 =
        (const float4*)(Rq + (((size_t)b * TT + s) * TT + t) * CC + h * FF);
    float ak = 0.f, aq = 0.f;
    for (int j = 0; j < 16; ++j) {
      float4 rk = rk4[j], rq = rq4[j];
      int f = j * 4;
      ak += qs[t][f] * rk.x + qs[t][f + 1] * rk.y + qs[t][f + 2] * rk.z +
            qs[t][f + 3] * rk.w;
      aq += ks[s][f] * rq.x + ks[s][f + 1] * rq.y + ks[s][f + 2] * rq.z +
            ks[s][f + 3] * rq.w;
    }
    float val = sc[t][s] + ak + SCALE * aq;
    sc[t][s] = (amask[b * TT + t] == amask[b * TT + s]) ? val : NEG_INF;
  }
  __syncthreads();

  // softmax over s (diagonal always allowed -> no empty rows)
  if (tid < TT) {
    int t = tid;
    float mx = NEG_INF;
    for (int s = 0; s < TT; ++s) mx = fmaxf(mx, sc[t][s]);
    float sum = 0.f;
    for (int s = 0; s < TT; ++s) {
      float e = __expf(sc[t][s] - mx);
      sc[t][s] = e;
      sum += e;
    }
    float inv = 1.f / sum;
    for (int s = 0; s < TT; ++s) sc[t][s] *= inv;
  }
  __syncthreads();

  // out1 = attn @ v : eight 16x16 tiles (2m x 4n), two per wave
  for (int j = 0; j < 2; ++j) {
    int st = wave * 2 + j;
    int sm = (st >> 2) * 16, sn = (st & 3) * 16;
    v8f acc = {0.f, 0.f, 0.f, 0.f, 0.f, 0.f, 0.f, 0.f};
    for (int kk = 0; kk < TT; kk += 4) {
      v2f a, bf;
      a.x = sc[sm + ml][kk + kl]; a.y = sc[sm + ml][kk + kl + 1];
      bf.x = vs[kk + kl][sn + ml]; bf.y = vs[kk + kl + 1][sn + ml];
      acc = wmma_f32_16x16x4(a, bf, acc);
    }
    for (int r = 0; r < 8; ++r) os[sm + r + mrow][sn + ml] = acc[r];
  }
  __syncthreads();

  // out2 = attn @ Rv, fused with final store
  {
    int t = tid & 31;
    int fb = (tid >> 5) * 16;
    float acc[16];
    for (int j = 0; j < 16; ++j) acc[j] = 0.f;
    for (int s = 0; s < TT; ++s) {
      float a = sc[t][s];
      const float4* rv4 = (const float4*)(Rv + (((size_t)b * TT + t) * TT + s) * CC +
                                          h * FF + fb);
      for (int j = 0; j < 4; ++j) {
        float4 r = rv4[j];
        acc[j * 4 + 0] += a * r.x; acc[j * 4 + 1] += a * r.y;
        acc[j * 4 + 2] += a * r.z; acc[j * 4 + 3] += a * r.w;
      }
    }
    float* op = out + (bd * TT + t) * CC + h * FF + fb;
    for (int j = 0; j < 16; ++j) op[j] = os[t][fb + j] + acc[j];
  }
}

// ---------------------------------------------------------------------------
// Host-side orchestration
// ---------------------------------------------------------------------------
extern "C" void kernel_launch(void* const* d_in, const int* in_sizes, int n_in,
                              void* d_out, int out_size, void* d_ws,
                              size_t ws_size, hipStream_t stream) {
  const float* x      = (const float*)d_in[0];
  const float* temb   = (const float*)d_in[1];
  const int*   fi     = (const int*)d_in[2];
  const int*   amask  = (const int*)d_in[3];
  const float* norm_w = (const float*)d_in[4];
  const float* norm_b = (const float*)d_in[5];
  const float* qkv_w  = (const float*)d_in[6];
  const float* qkv_b  = (const float*)d_in[7];
  const float* proj_w = (const float*)d_in[8];
  const float* proj_b = (const float*)d_in[9];

  float* ws = (float*)d_ws;
  // workspace layout (floats)
  float* xt       = ws;                       // 16,777,216  [B*D,T,C]
  float* qkvbuf   = ws + 16777216ull;         // 50,331,648  [B*D,T,3C]
  float* Rbase    = ws + 67108864ull;         // 3 x 1,048,576 (Rq,Rk,Rv)
  float* silu_emb = ws + 70254592ull;         // 1,048,576
  float* tproj    = ws + 71303168ull;         // 32,768
  float* attn_out = ws + 71335936ull;         // 16,777,216  [B*D,T,C]

  // 1) GroupNorm + transpose
  groupnorm_kernel<<<BB * DD, 256, 0, stream>>>(x, norm_w, norm_b, xt);

  // 2) RPE bias tensors: p = 0:q, 1:k, 2:v  (inputs at 10 + 6*p)
  for (int p = 0; p < 3; ++p) {
    const float* dw = (const float*)d_in[10 + 6 * p + 0];
    const float* db = (const float*)d_in[10 + 6 * p + 1];
    const float* tw = (const float*)d_in[10 + 6 * p + 2];
    const float* tb = (const float*)d_in[10 + 6 * p + 3];
    const float* ow = (const float*)d_in[10 + 6 * p + 4];
    const float* ob = (const float*)d_in[10 + 6 * p + 5];
    float* Rp = Rbase + (size_t)p * 1048576ull;
    // tproj[B*T, C] = temb[B*T, TDIM] @ tw^T + tb
    gemm_wmma_kernel<<<dim3(CC / 64, (BB * TT) / 64), 256, 0, stream>>>(
        temb, tw, tb, tproj, BB * TT, CC, TDIM, nullptr, 0, 0);
    rpe_emb_kernel<<<(BB * TT * TT * CC) / 256, 256, 0, stream>>>(
        tproj, dw, db, fi, silu_emb);
    // Rp[B*T*T, C] = silu_emb @ ow^T + ob
    gemm_wmma_kernel<<<dim3(CC / 64, (BB * TT * TT) / 64), 256, 0, stream>>>(
        silu_emb, ow, ob, Rp, BB * TT * TT, CC, CC, nullptr, 0, 0);
  }

  // 3) QKV projection: [32768, 1536] = xt @ qkv_w^T + qkv_b
  gemm_wmma_kernel<<<dim3((3 * CC) / 64, (BB * DD * TT) / 64), 256, 0, stream>>>(
      xt, qkv_w, qkv_b, qkvbuf, BB * DD * TT, 3 * CC, CC, nullptr, 0, 0);

  // 4) Attention core per (b,d,h)
  attn_kernel<<<BB * DD * HH, 128, 0, stream>>>(
      qkvbuf, /*Rk=*/Rbase + 1048576ull, /*Rq=*/Rbase,
      /*Rv=*/Rbase + 2097152ull, amask, attn_out);

  // 5) proj GEMM fused with residual (+xt) and transposed store into d_out
  gemm_wmma_kernel<<<dim3(CC / 64, (BB * DD * TT) / 64), 256, 0, stream>>>(
      attn_out, proj_w, proj_b, (float*)d_out, BB * DD * TT, CC, CC,
      /*resid=*/xt, /*mode=*/1, /*Tdim=*/TT);
}